// RSSM_2765958938713
// MI455X (gfx1250) — compile-verified
//
#include <hip/hip_runtime.h>

// RSSM fused persistent scan for gfx1250 (MI455X), v3.
// - 128 blocks x 64 threads: TWO wave32s cooperate on one 16-row batch tile,
//   splitting column tiles (gate triplets stay wave-local), halving the serial
//   per-step critical path and doubling device wave occupancy vs v2.
// - Weights pre-swizzled in LDS (B-fragment order): ds_load_b64 -> wmma.
// - Hardware transcendentals (v_exp_f32 / v_rcp_f32) for sigmoid/tanh/softmax:
//   TRANS ops co-execute with WMMA on CDNA5, and cut VALU count ~10x vs ocml.
// - fp32 WMMA (v_wmma_f32_16x16x4_f32) keeps native fp32 numerics.

#define TT 256
#define NB 2048
#define HH 32
#define ZZ 64
#define ZSs 8
#define ZCc 8
#define XX 10
#define AA 4
#define NTHR 64   // 2 waves per block

// State row layout (stride SSTR floats):
//  [0..31] h | [32..41] e_t | [42..43] zero pad | [44..107] z_prev | [108..111] a_prev
#define SSTR 112
#define COL_H 0
#define COL_E 32
#define COL_Z 44
#define COL_A 108

// Packed-weight LDS offsets (floats). Each chunk = 64 floats = 32 lanes x K-pair.
#define PW_IH  0                    // 6 tiles x 17 kc
#define PW_HH  (PW_IH + 6*17*64)    // 6 x 8
#define PW_ENC (PW_HH + 6*8*64)     // 4 x 11
#define PW_DEC (PW_ENC + 4*11*64)   // 1 x 24
#define PW_DYN (PW_DEC + 24*64)     // 4 x 8
#define PW_TOT (PW_DYN + 4*8*64)    // 16000 floats = 64000 B

#define B_ENC 0
#define B_IH  64
#define B_HH  160
#define B_DEC 256
#define B_DYN 272
#define B_REW 336
#define B_CON 368
#define B_CB  400
#define B_TOT 408

typedef float v2f __attribute__((ext_vector_type(2)));
typedef float v8f __attribute__((ext_vector_type(8)));

// ---- hardware transcendentals (v_exp_f32 computes 2^x; ~1 ulp) ----
__device__ __forceinline__ float fexp(float x) {
  return __builtin_amdgcn_exp2f(x * 1.44269504088896340736f);
}
__device__ __forceinline__ float frcp(float x) { return __builtin_amdgcn_rcpf(x); }
__device__ __forceinline__ float sigm(float x) { return frcp(1.0f + fexp(-x)); }
__device__ __forceinline__ float ftanh(float x) {
  const float t = fexp(-2.0f * fabsf(x));        // in (0,1]: no inf-inf
  return copysignf((1.0f - t) * frcp(1.0f + t), x);
}

// A fragment (16x4 fp32) from LDS [16][stride] at column k0.
// lane l: row M=l%16, K = k0 + v + 2*(l/16)  (ds_load_b64)
__device__ __forceinline__ v2f frag_a(const float* base, int stride, int k0, int lane) {
  const float* p = base + (lane & 15) * stride + k0 + ((lane >> 4) << 1);
  v2f r; r.x = p[0]; r.y = p[1];
  return r;
}

// B fragment from pre-swizzled LDS weights: chunk*64 + lane*2 (ds_load_b64).
__device__ __forceinline__ v2f frag_bp(const float* pw, int chunk, int lane) {
  const float* p = pw + chunk * 64 + lane * 2;
  v2f r; r.x = p[0]; r.y = p[1];
  return r;
}

// Pack W[nmax][ws] (valid K = kmax) into B-fragment order (all 64 threads help).
__device__ __forceinline__ void pack_w(float* dst, const float* __restrict__ W,
                                       int ws, int CT, int KC, int nmax, int kmax,
                                       int tid) {
  const int total = CT * KC * 32;
  for (int idx = tid; idx < total; idx += NTHR) {
    const int l = idx & 31;
    const int chunk = idx >> 5;
    const int kc = chunk % KC;
    const int ct = chunk / KC;
    const int n = ct * 16 + (l & 15);
    const int k = kc * 4 + ((l >> 4) << 1);
    float x0 = 0.0f, x1 = 0.0f;
    if (n < nmax) {
      if (k < kmax)     x0 = W[n * ws + k];
      if (k + 1 < kmax) x1 = W[n * ws + k + 1];
    }
    dst[chunk * 64 + l * 2 + 0] = x0;
    dst[chunk * 64 + l * 2 + 1] = x1;
  }
}

// D fragment (row = v + 8*(lane/16), col = ct*16 + lane%16) -> [16][64] LDS tile.
__device__ __forceinline__ void frag_to_tile(float (*dstT)[64], int ct, v8f acc, int lane) {
  const int col = ct * 16 + (lane & 15);
  const int r0 = (lane >> 4) << 3;
#pragma unroll
  for (int v = 0; v < 8; ++v) dstT[r0 + v][col] = acc[v];
}

// Coalesced [16][64] LDS tile -> global rows (row stride 64), float4 stores.
__device__ __forceinline__ void tile_to_global(const float (*src)[64], float* dst, int tid) {
  for (int idx = tid; idx < 16 * 16; idx += NTHR) {
    const int row = idx >> 4, q = idx & 15;
    const float4 v = *(const float4*)&src[row][q * 4];
    *(float4*)&dst[(size_t)row * ZZ + q * 4] = v;
  }
}

__global__ void __launch_bounds__(NTHR)
rssm_fused_scan(const float* __restrict__ x, const float* __restrict__ a,
                const float* __restrict__ h0, const float* __restrict__ gum,
                const float* __restrict__ encW, const float* __restrict__ encB,
                const float* __restrict__ Wih, const float* __restrict__ Whh,
                const float* __restrict__ bih, const float* __restrict__ bhh,
                const float* __restrict__ decW, const float* __restrict__ decB,
                const float* __restrict__ dynW, const float* __restrict__ dynB,
                const float* __restrict__ rewW, const float* __restrict__ conW,
                const float* __restrict__ conB,
                float* __restrict__ o_x, float* __restrict__ o_r,
                float* __restrict__ o_c, float* __restrict__ o_zp,
                float* __restrict__ o_zq) {
  __shared__ float sW[PW_TOT];     // 64000 B packed weights (shared by both waves)
  __shared__ float sB[B_TOT];
  __shared__ float S[16][SSTR];    // recurrent state tile
  __shared__ float ZL[16][ZZ];     // prior logits tile
  __shared__ float OT[16][ZZ];     // dyn-head staging tile

  const int tid = threadIdx.x;
  const int wave = tid >> 5;       // 0 or 1: column-tile owner
  const int lane = tid & 31;
  const int nbase = blockIdx.x * 16;

  // ---- one-time: pack weights/biases, load h0, zero pads ----
  pack_w(sW + PW_IH,  Wih,  68, 6, 17, 96, 68, tid);
  pack_w(sW + PW_HH,  Whh,  32, 6,  8, 96, 32, tid);
  pack_w(sW + PW_ENC, encW, 42, 4, 11, 64, 42, tid);
  pack_w(sW + PW_DEC, decW, 96, 1, 24, 10, 96, tid);
  pack_w(sW + PW_DYN, dynW, 32, 4,  8, 64, 32, tid);
  for (int i = tid; i < 64; i += NTHR) sB[B_ENC + i] = encB[i];
  for (int i = tid; i < 96; i += NTHR) sB[B_IH + i] = bih[i];
  for (int i = tid; i < 96; i += NTHR) sB[B_HH + i] = bhh[i];
  for (int i = tid; i < 16; i += NTHR) sB[B_DEC + i] = (i < XX) ? decB[i] : 0.0f;
  for (int i = tid; i < 64; i += NTHR) sB[B_DYN + i] = dynB[i];
  for (int i = tid; i < 32; i += NTHR) sB[B_REW + i] = rewW[i];
  for (int i = tid; i < 32; i += NTHR) sB[B_CON + i] = conW[i];
  if (tid == 0) sB[B_CB] = conB[0];
  for (int i = tid; i < 16 * SSTR; i += NTHR) (&S[0][0])[i] = 0.0f;
  for (int i = tid; i < 16 * HH; i += NTHR)
    S[i >> 5][COL_H + (i & 31)] = h0[(size_t)(nbase + (i >> 5)) * HH + (i & 31)];
  __syncthreads();

  const int fcol = lane & 15;
  const int fr0 = (lane >> 4) << 3;

  for (int t = 0; t < TT; ++t) {
    // ---- stage e_t (and a_{t-1}) ----
    for (int i = tid; i < 16 * XX; i += NTHR)
      S[i / XX][COL_E + i % XX] = x[(size_t)(t * NB + nbase + i / XX) * XX + i % XX];
    if (t > 0) {
      for (int i = tid; i < 16 * AA; i += NTHR)
        S[i >> 2][COL_A + (i & 3)] = a[(size_t)((t - 1) * NB + nbase + (i >> 2)) * AA + (i & 3)];
    }
    __syncthreads();  // staging -> GEMM reads

    if (t > 0) {
      // wave w owns gate tiles {w, w+2, w+4}: r/z/n triplet for h-cols [16w,16w+16)
      v8f ghf[3], gif[3];
#pragma unroll
      for (int k = 0; k < 3; ++k) {
        const int ct = wave + 2 * k;
        v8f acc = {};
#pragma unroll
        for (int kc = 0; kc < 8; ++kc) {
          const v2f av = frag_a(&S[0][COL_H], SSTR, kc * 4, lane);
          const v2f bv = frag_bp(sW + PW_HH, ct * 8 + kc, lane);
          acc = __builtin_amdgcn_wmma_f32_16x16x4_f32(false, av, false, bv,
                                                      (short)0, acc, false, false);
        }
        ghf[k] = acc;
      }
#pragma unroll
      for (int k = 0; k < 3; ++k) {
        const int ct = wave + 2 * k;
        v8f acc = {};
#pragma unroll
        for (int kc = 0; kc < 17; ++kc) {
          const v2f av = frag_a(&S[0][COL_Z], SSTR, kc * 4, lane);
          const v2f bv = frag_bp(sW + PW_IH, ct * 17 + kc, lane);
          acc = __builtin_amdgcn_wmma_f32_16x16x4_f32(false, av, false, bv,
                                                      (short)0, acc, false, false);
        }
        gif[k] = acc;
      }
      __syncthreads();  // all old-h reads done before h is overwritten

      // fused GRU gates on fragments; this wave updates h-cols [16*wave, 16*wave+16)
      {
        const int j = wave * 16 + fcol;
        const float bi_r = sB[B_IH + j],      bh_r = sB[B_HH + j];
        const float bi_z = sB[B_IH + 32 + j], bh_z = sB[B_HH + 32 + j];
        const float bi_n = sB[B_IH + 64 + j], bh_n = sB[B_HH + 64 + j];
#pragma unroll
        for (int v = 0; v < 8; ++v) {
          const float rg = sigm(gif[0][v] + bi_r + ghf[0][v] + bh_r);
          const float zg = sigm(gif[1][v] + bi_z + ghf[1][v] + bh_z);
          const float ng = ftanh(gif[2][v] + bi_n + rg * (ghf[2][v] + bh_n));
          const float hv = S[fr0 + v][COL_H + j];
          S[fr0 + v][COL_H + j] = (1.0f - zg) * ng + zg * hv;
        }
      }
      __syncthreads();  // full new h visible
    }

    // ---- encoder: zl = [h|e] @ encW^T + encB ; wave w does tiles {w, w+2} ----
#pragma unroll
    for (int k = 0; k < 2; ++k) {
      const int ct = wave + 2 * k;
      v8f acc = {};
#pragma unroll
      for (int kc = 0; kc < 11; ++kc) {
        const v2f av = frag_a(&S[0][COL_H], SSTR, kc * 4, lane);
        const v2f bv = frag_bp(sW + PW_ENC, ct * 11 + kc, lane);
        acc = __builtin_amdgcn_wmma_f32_16x16x4_f32(false, av, false, bv,
                                                    (short)0, acc, false, false);
      }
      const float bias = sB[B_ENC + ct * 16 + fcol];
#pragma unroll
      for (int v = 0; v < 8; ++v) acc[v] += bias;
      frag_to_tile(ZL, ct, acc, lane);
    }
    __syncthreads();  // ZL complete
    tile_to_global(ZL, o_zp + (size_t)(t * NB + nbase) * ZZ, tid);  // prior logits

    // ---- straight-through gumbel-argmax sample -> z ----
    for (int i = tid; i < 16 * ZSs; i += NTHR) {
      const int r = i >> 3, s = i & 7;
      const float4 la = *(const float4*)&ZL[r][s * 8];
      const float4 lb = *(const float4*)&ZL[r][s * 8 + 4];
      const float4 ga = *(const float4*)&gum[(size_t)(t * NB + nbase + r) * ZZ + s * 8];
      const float4 gb = *(const float4*)&gum[(size_t)(t * NB + nbase + r) * ZZ + s * 8 + 4];
      float l[8] = {la.x, la.y, la.z, la.w, lb.x, lb.y, lb.z, lb.w};
      float g[8] = {ga.x, ga.y, ga.z, ga.w, gb.x, gb.y, gb.z, gb.w};
      int best = 0;
      float bestv = l[0] + g[0], mx = l[0];
#pragma unroll
      for (int c2 = 1; c2 < ZCc; ++c2) {
        const float v = l[c2] + g[c2];
        if (v > bestv) { bestv = v; best = c2; }  // first-max tie-break (jnp.argmax)
        if (l[c2] > mx) mx = l[c2];
      }
      float ex[ZCc];
      float se = 0.0f;
#pragma unroll
      for (int c2 = 0; c2 < ZCc; ++c2) { ex[c2] = fexp(l[c2] - mx); se += ex[c2]; }
      const float inv = frcp(se);
#pragma unroll
      for (int c2 = 0; c2 < ZCc; ++c2) {
        const float p = ex[c2] * inv;
        const float oh = (c2 == best) ? 1.0f : 0.0f;
        S[r][COL_Z + s * 8 + c2] = p + (oh - p);   // == probs + sg(onehot - probs)
      }
    }
    __syncthreads();  // z visible to heads (and next-step GRU)

    if (wave == 0) {
      // ---- decoder head: [h|z] @ decW^T + decB (K split 32+64) ----
      v8f acc = {};
#pragma unroll
      for (int kc = 0; kc < 24; ++kc) {
        const v2f av = (kc < 8) ? frag_a(&S[0][COL_H], SSTR, kc * 4, lane)
                                : frag_a(&S[0][COL_Z], SSTR, (kc - 8) * 4, lane);
        const v2f bv = frag_bp(sW + PW_DEC, kc, lane);
        acc = __builtin_amdgcn_wmma_f32_16x16x4_f32(false, av, false, bv,
                                                    (short)0, acc, false, false);
      }
      if (fcol < XX) {
        const float bias = sB[B_DEC + fcol];
#pragma unroll
        for (int v = 0; v < 8; ++v)
          o_x[(size_t)(t * NB + nbase + fr0 + v) * XX + fcol] = acc[v] + bias;
      }
      // ---- reward / continue heads (1-wide): VALU dots ----
      {
        const int r = lane & 15;
        const int which = lane >> 4;
        const float* w = sB + (which ? B_CON : B_REW);
        float accv = 0.0f;
#pragma unroll
        for (int kk = 0; kk < HH; ++kk) accv += S[r][COL_H + kk] * w[kk];
        if (which)
          o_c[(size_t)t * NB + nbase + r] = accv + sB[B_CB];
        else
          o_r[(size_t)t * NB + nbase + r] = accv;
      }
    } else {
      // ---- dynamics head: h @ dynW^T + dynB (all 4 tiles) ----
#pragma unroll
      for (int ct = 0; ct < 4; ++ct) {
        v8f acc = {};
#pragma unroll
        for (int kc = 0; kc < 8; ++kc) {
          const v2f av = frag_a(&S[0][COL_H], SSTR, kc * 4, lane);
          const v2f bv = frag_bp(sW + PW_DYN, ct * 8 + kc, lane);
          acc = __builtin_amdgcn_wmma_f32_16x16x4_f32(false, av, false, bv,
                                                      (short)0, acc, false, false);
        }
        const float bias = sB[B_DYN + ct * 16 + fcol];
#pragma unroll
        for (int v = 0; v < 8; ++v) acc[v] += bias;
        frag_to_tile(OT, ct, acc, lane);
      }
    }
    __syncthreads();  // OT complete
    tile_to_global(OT, o_zq + (size_t)(t * NB + nbase) * ZZ, tid);  // post logits
    // (no extra barrier needed: next writers of S-e/a, ZL, OT sit behind the
    //  top-of-loop and enc/head barriers of iteration t+1)
  }
}

extern "C" void kernel_launch(void* const* d_in, const int* in_sizes, int n_in,
                              void* d_out, int out_size, void* d_ws, size_t ws_size,
                              hipStream_t stream) {
  (void)in_sizes; (void)n_in; (void)out_size; (void)d_ws; (void)ws_size;
  const float* x    = (const float*)d_in[0];
  const float* a    = (const float*)d_in[1];
  const float* h0   = (const float*)d_in[2];
  const float* gum  = (const float*)d_in[3];
  const float* encW = (const float*)d_in[4];
  const float* encB = (const float*)d_in[5];
  const float* Wih  = (const float*)d_in[6];
  const float* Whh  = (const float*)d_in[7];
  const float* bih  = (const float*)d_in[8];
  const float* bhh  = (const float*)d_in[9];
  const float* decW = (const float*)d_in[10];
  const float* decB = (const float*)d_in[11];
  const float* dynW = (const float*)d_in[12];
  const float* dynB = (const float*)d_in[13];
  const float* rewW = (const float*)d_in[14];
  const float* conW = (const float*)d_in[15];
  const float* conB = (const float*)d_in[16];

  float* out  = (float*)d_out;
  float* o_x  = out;                          // T*N*10
  float* o_r  = o_x + (size_t)TT * NB * XX;   // T*N
  float* o_c  = o_r + (size_t)TT * NB;        // T*N
  float* o_zp = o_c + (size_t)TT * NB;        // T*N*64
  float* o_zq = o_zp + (size_t)TT * NB * ZZ;  // T*N*64

  rssm_fused_scan<<<dim3(NB / 16), dim3(NTHR), 0, stream>>>(
      x, a, h0, gum, encW, encB, Wih, Whh, bih, bhh,
      decW, decB, dynW, dynB, rewW, conW, conB,
      o_x, o_r, o_c, o_zp, o_zq);
}